// ContinuousWaveletTransform_6622839570940
// MI455X (gfx1250) — compile-verified
//
#include <hip/hip_runtime.h>

// CWT = 129-tap FIR, 4 filters, along W of a (8,16,128,1024) f32 tensor.
// V_WMMA_F32_16X16X4_F32 with M = (scale, w-subgroup) folding:
//   A[m,k'] = bank[s, k'-16g]  (m = s*4+g, zero outside taps)
//   B[k',n] = sig[tile_base + n + k' - 64]
//   D[m,n]  = out[s, tile_base + 16g + n]
// Signal row staged into LDS via global_load_async_to_lds_b128 (ASYNCcnt).

typedef __attribute__((ext_vector_type(2))) float v2f;
typedef __attribute__((ext_vector_type(8))) float v8f;

#define NTAPS   129     // 2*4*max(SCALES)+1
#define KTOT    180     // 129 taps + 48 (3*16 M-fold shift) -> 177, pad to mult of 4
#define NCHUNK  (KTOT / 4)
#define MROWS   16      // m = s*4 + g  (4 scales x 4 w-groups of 16)
#define HALO    64
#define WLEN    1024
#define SIG_LDS 1216    // covers signal indices [-64, 1152), zero-padded

// ---------------------------------------------------------------------------
// Init: shifted-tap matrix A[m][k'] = bank[s][k' - 16*g] (0 outside)
// bank[s][k] = exp(-0.5*((k-64)/sc)^2) * cos(5*(k-64)/sc) / sqrt(sc), sc = 2<<s
// ---------------------------------------------------------------------------
__global__ void cwt_bank_init(float* __restrict__ At) {
    int idx = blockIdx.x * blockDim.x + threadIdx.x;
    if (idx >= MROWS * KTOT) return;
    int m  = idx / KTOT;
    int kp = idx % KTOT;
    int s  = m >> 2;
    int g  = m & 3;
    int k  = kp - 16 * g;                 // tap index
    float v = 0.0f;
    if (k >= 0 && k < NTAPS) {
        float sc = 2.0f * (float)(1 << s);
        float ts = (float)(k - 64) / sc;
        v = expf(-0.5f * ts * ts) * cosf(5.0f * ts) * rsqrtf(sc);
    }
    At[idx] = v;
}

// ---------------------------------------------------------------------------
// Main: one block per row (16384 rows), 4 waves. Each wave owns 256 outputs
// (4 supertiles of 64). Loop interchange: A fragment loaded once per K-chunk,
// reused by 4 WMMAs into 4 independent accumulators.
// ---------------------------------------------------------------------------
__global__ void __launch_bounds__(128) cwt_wmma(const float* __restrict__ x,
                                                const float* __restrict__ At,
                                                float* __restrict__ out) {
    __shared__ __align__(16) float sAt[MROWS * KTOT];   // 11.25 KB
    __shared__ __align__(16) float sSig[SIG_LDS];       // 4.75 KB, sSig[i] = row[i-64]

    const int tid = threadIdx.x;
    const int r   = blockIdx.x;           // row id = (b*16+c)*128 + h
    const int bc  = r >> 7;
    const int h   = r & 127;
    const float* __restrict__ row = x + (size_t)r * WLEN;

    // Tap table: plain load (L2-resident, 11.25 KB).
    for (int i = tid; i < MROWS * KTOT; i += 128) sAt[i] = At[i];

    // Halo zero-fill ([1088,1216) disjoint from async interior; [64,128) overlaps it).
    sSig[tid] = 0.0f;                     // indices [0,128) -> covers [0,64) halo (+64 extra, overwritten)
    sSig[1088 + tid] = 0.0f;              // indices [1088,1216): beyond signal

    // Interior [64,1088) <- row[0..1024): async memory->LDS, 2x b128 per lane.
    {
        const int si0 = tid * 4;                 // floats 0..511
        const int si1 = 512 + tid * 4;           // floats 512..1023
        unsigned long long ga0 = (unsigned long long)(unsigned long long)(uintptr_t)(row + si0);
        unsigned long long ga1 = (unsigned long long)(unsigned long long)(uintptr_t)(row + si1);
        unsigned la0 = (unsigned)(uintptr_t)&sSig[HALO + si0];   // LDS addr = low 32 bits
        unsigned la1 = (unsigned)(uintptr_t)&sSig[HALO + si1];
        asm volatile("global_load_async_to_lds_b128 %0, %1, off"
                     :: "v"(la0), "v"(ga0) : "memory");
        asm volatile("global_load_async_to_lds_b128 %0, %1, off"
                     :: "v"(la1), "v"(ga1) : "memory");
        asm volatile("s_wait_asynccnt 0x0" ::: "memory");
    }
    __syncthreads();

    const int lane = tid & 31;            // wave32
    const int wv   = tid >> 5;
    const int hh   = lane >> 4;           // lane half: K pair {0,1} vs {2,3}
    const int ln   = lane & 15;           // A row m and B column n
    const int koff = 2 * hh;

    const float* __restrict__ ap = &sAt[ln * KTOT];
    const float* __restrict__ sp = &sSig[wv * 256 + ln];  // +64 halo cancels -64 tap offset

    v8f acc0 = {}, acc1 = {}, acc2 = {}, acc3 = {};

    for (int c = 0; c < NCHUNK; ++c) {
        const int k0 = 4 * c + koff;
        v2f a, b0, b1, b2, b3;
        a.x  = ap[k0];        a.y  = ap[k0 + 1];       // 8B-aligned pair -> b64
        b0.x = sp[k0];        b0.y = sp[k0 + 1];
        b1.x = sp[64 + k0];   b1.y = sp[64 + k0 + 1];
        b2.x = sp[128 + k0];  b2.y = sp[128 + k0 + 1];
        b3.x = sp[192 + k0];  b3.y = sp[192 + k0 + 1];
        acc0 = __builtin_amdgcn_wmma_f32_16x16x4_f32(false, a, false, b0, (short)0, acc0, false, false);
        acc1 = __builtin_amdgcn_wmma_f32_16x16x4_f32(false, a, false, b1, (short)0, acc1, false, false);
        acc2 = __builtin_amdgcn_wmma_f32_16x16x4_f32(false, a, false, b2, (short)0, acc2, false, false);
        acc3 = __builtin_amdgcn_wmma_f32_16x16x4_f32(false, a, false, b3, (short)0, acc3, false, false);
    }

    // C/D layout: VGPR j holds D[m][n], m = j + 8*hh, n = ln.
    // m = s*4 + g  ->  w = base + g*16 + n   (16-lane-contiguous 64B stores)
    const size_t obase = (size_t)bc * (4u * 128u * 1024u) + (size_t)h * 1024u;
    {
        v8f accs[4] = {acc0, acc1, acc2, acc3};
        #pragma unroll
        for (int st = 0; st < 4; ++st) {
            const int base = wv * 256 + st * 64;
            #pragma unroll
            for (int j = 0; j < 8; ++j) {
                const int m = j + 8 * hh;
                const int s = m >> 2;
                const int g = m & 3;
                out[obase + (size_t)s * (128u * 1024u) + (unsigned)(base + g * 16 + ln)] =
                    accs[st][j];
            }
        }
    }
}

// ---------------------------------------------------------------------------
extern "C" void kernel_launch(void* const* d_in, const int* in_sizes, int n_in,
                              void* d_out, int out_size, void* d_ws, size_t ws_size,
                              hipStream_t stream) {
    (void)in_sizes; (void)n_in; (void)out_size; (void)ws_size;
    const float* x  = (const float*)d_in[0];
    float*       o  = (float*)d_out;
    float*       At = (float*)d_ws;       // 16*180 floats = 11.25 KB scratch

    cwt_bank_init<<<(MROWS * KTOT + 255) / 256, 256, 0, stream>>>(At);
    cwt_wmma<<<16384, 128, 0, stream>>>(x, At, o);
}